// MeanAggregator_841813590039
// MI455X (gfx1250) — compile-verified
//
#include <hip/hip_runtime.h>
#include <stdint.h>

// MeanAggregator: out[b,:] = mean_k table[neighs[b,k], :]
// B=50000, K=32 neighbors, D=128 fp32 features.
//
// Memory/latency-bound random gather. CDNA5 path: async global->LDS row
// staging (GLOBAL_LOAD_ASYNC_TO_LDS_B128, GVS mode: constant 64-bit SADDR,
// signed 32-bit VADDR carries the row offset, IOFFSET immediate selects
// the LDS row). One wave per block so the neighbor-index block is
// block-uniform -> indices load straight into SGPRs via SMEM (no VMEM,
// no readlanes). Zero destination VGPRs -> 16 KB in flight per wave on
// ASYNCcnt; split s_wait_asynccnt overlaps accumulation with the tail
// fetches; NT output stores keep the 192MB L2 for the 256MB table
// (~3.2x row reuse per batch).

#define KNBR   32
#define DFEAT  128
#define LANES  32         // wave32; one wave per block

typedef __attribute__((ext_vector_type(4))) float v4f;

__global__ __launch_bounds__(LANES)
void mean_aggregate_kernel(const long long* __restrict__ neighs,
                           const float* __restrict__ table,
                           float* __restrict__ out,
                           int batch) {
    // One 32x128 f32 staging tile (16 KB) -> 20 blocks/WGP on 320 KB LDS.
    __shared__ v4f buf[KNBR][DFEAT / 4];

    const int lane = threadIdx.x;      // 0..31
    const int node = blockIdx.x;       // block-uniform -> scalar control flow
    if (node >= batch) return;

    // Block-uniform index block: plain loads from a uniform address become
    // s_load (SMEM) into SGPRs, merged into wide s_load_bN chunks. Hoisted
    // before the first asm "memory" clobber so SMEM selection survives.
    const long long* __restrict__ idxp = neighs + (size_t)node * KNBR;
    int idxv[KNBR];
    #pragma unroll
    for (int k = 0; k < KNBR; ++k) idxv[k] = (int)idxp[k];

    const uint64_t gbase  = (uint64_t)(uintptr_t)table;   // constant SADDR
    // Low 32 bits of the flat address of a __shared__ object == wave-relative
    // LDS byte address on gfx1250 (LDS aperture lives in addr[63:32]).
    const uint32_t lane16 = (uint32_t)lane * 16u;
    const uint32_t ldst   = (uint32_t)(uintptr_t)(&buf[0][0]) + lane16;

    // Stage all 32 neighbor rows. GVS: mem = SADDR + VADDR(i32) + IOFFSET,
    // LDS = VDST + IOFFSET. SADDR stays gbase for every row; the per-row
    // scalar term (idx*512 - k*512) is pure SALU and rides in the signed
    // 32-bit VADDR (one v_add with lane16); IOFFSET k*512 picks the LDS row
    // (it is added to BOTH addresses, hence the -k*512 compensation).
    // One instruction moves a full 512B row (32 lanes x b128) with no
    // destination VGPRs -> 16 KB in flight per wave on ASYNCcnt.
#define ISSUE_ROW(kk)                                                        \
    {                                                                        \
        const int soff = (idxv[kk] << 9) - (kk) * 512;                       \
        const uint32_t vaddr = lane16 + (uint32_t)soff;                      \
        asm volatile("global_load_async_to_lds_b128 %0, %1, %2 offset:%c3"   \
                     :: "v"(ldst), "v"(vaddr), "s"(gbase),                   \
                        "i"((kk) * 512)                                      \
                     : "memory");                                            \
    }

    ISSUE_ROW(0)  ISSUE_ROW(1)  ISSUE_ROW(2)  ISSUE_ROW(3)
    ISSUE_ROW(4)  ISSUE_ROW(5)  ISSUE_ROW(6)  ISSUE_ROW(7)
    ISSUE_ROW(8)  ISSUE_ROW(9)  ISSUE_ROW(10) ISSUE_ROW(11)
    ISSUE_ROW(12) ISSUE_ROW(13) ISSUE_ROW(14) ISSUE_ROW(15)
    ISSUE_ROW(16) ISSUE_ROW(17) ISSUE_ROW(18) ISSUE_ROW(19)
    ISSUE_ROW(20) ISSUE_ROW(21) ISSUE_ROW(22) ISSUE_ROW(23)
    ISSUE_ROW(24) ISSUE_ROW(25) ISSUE_ROW(26) ISSUE_ROW(27)
    ISSUE_ROW(28) ISSUE_ROW(29) ISSUE_ROW(30) ISSUE_ROW(31)
#undef ISSUE_ROW

    // Async loads return "done" in order: asynccnt<=16 => rows 0..15 landed.
    // Overlap their accumulation with the in-flight tail fetches.
    v4f acc = (v4f)0.0f;
    asm volatile("s_wait_asynccnt 16" ::: "memory");
    #pragma unroll
    for (int k = 0; k < KNBR / 2; ++k) {
        acc += buf[k][lane];             // ds_load_b128
    }
    asm volatile("s_wait_asynccnt 0" ::: "memory");
    #pragma unroll
    for (int k = KNBR / 2; k < KNBR; ++k) {
        acc += buf[k][lane];
    }

    acc *= (1.0f / (float)KNBR);

    // Output is write-once -> non-temporal b128 store, keeps L2 for the table.
    v4f* dst = (v4f*)&out[(long long)node * DFEAT + lane * 4];
    __builtin_nontemporal_store(acc, dst);
}

extern "C" void kernel_launch(void* const* d_in, const int* in_sizes, int n_in,
                              void* d_out, int out_size, void* d_ws, size_t ws_size,
                              hipStream_t stream) {
    const long long* neighs = (const long long*)d_in[0]; // [B, 32] int64
    const float*     table  = (const float*)d_in[1];     // [500000, 128] f32
    float*           out    = (float*)d_out;             // [B, 128] f32

    const int batch = in_sizes[0] / KNBR;

    dim3 block(LANES);
    dim3 grid(batch);
    mean_aggregate_kernel<<<grid, block, 0, stream>>>(neighs, table, out, batch);
}